// GCNModel_78821239816828
// MI455X (gfx1250) — compile-verified
//
#include <hip/hip_runtime.h>
#include <math.h>

#define N_NODES   100000
#define N_EDGES   1600000
#define N_GRAPHS  4096
#define IN_FEATS  64
#define HIDDEN    128

typedef __attribute__((ext_vector_type(2))) float v2f;
typedef __attribute__((ext_vector_type(8))) float v8f;

__device__ __forceinline__ float selu_f(float x) {
    const float scale = 1.0507009873554804934193349852946f;
    const float alpha = 1.6732632423543772848170429916717f;
    return scale * (x > 0.0f ? x : alpha * (expf(x) - 1.0f));
}

// ---------------------------------------------------------------- utilities
__global__ void zero_kernel(float* __restrict__ p, long n) {
    for (long i = blockIdx.x * (long)blockDim.x + threadIdx.x; i < n;
         i += (long)gridDim.x * blockDim.x)
        p[i] = 0.0f;
}

__global__ void degree_kernel(const int* __restrict__ src, const int* __restrict__ dst,
                              float* __restrict__ degOut, float* __restrict__ degIn, int E) {
    for (int i = blockIdx.x * blockDim.x + threadIdx.x; i < E;
         i += gridDim.x * blockDim.x) {
        atomicAdd(&degOut[src[i]], 1.0f);
        atomicAdd(&degIn[dst[i]], 1.0f);
    }
}

__global__ void finalize_norm_kernel(float* __restrict__ deg, int n) {
    for (int i = blockIdx.x * blockDim.x + threadIdx.x; i < n;
         i += gridDim.x * blockDim.x) {
        float d = deg[i];
        d = d < 1.0f ? 1.0f : d;
        deg[i] = 1.0f / sqrtf(d);
    }
}

// ------------------------------------------------ edge gather + segment sum
// agg[dst] += x[src] * normOut[src]   (one thread = one edge x 4 features)
__global__ void edge_agg_kernel(const float* __restrict__ x, const float* __restrict__ normOut,
                                const int* __restrict__ src, const int* __restrict__ dst,
                                float* __restrict__ agg, int E, int F) {
    const int chunks = F >> 2;
    const long total = (long)E * chunks;
    for (long i = blockIdx.x * (long)blockDim.x + threadIdx.x; i < total;
         i += (long)gridDim.x * blockDim.x) {
        const int e = (int)(i / chunks);
        const int c = (int)(i - (long)e * chunks);
        const int s = src[e];
        const int d = dst[e];
        const float w = normOut[s];
        const float4 v = reinterpret_cast<const float4*>(x)[(long)s * chunks + c];
        float* o = agg + (long)d * F + (c << 2);
        atomicAdd(o + 0, v.x * w);
        atomicAdd(o + 1, v.y * w);
        atomicAdd(o + 2, v.z * w);
        atomicAdd(o + 3, v.w * w);
    }
}

// ------------------------------------------------------------ WMMA f32 GEMM
// Out[M,N] = selu?( (A[M,K] * rowScale[M]) @ W[K,N] + bias[N] )
// one wave -> one 16x16 tile; blockDim = (N/16)*32; grid = M/16
__global__ void gemm_wmma_kernel(const float* __restrict__ A,
                                 const float* __restrict__ rowScale,  // may be null
                                 const float* __restrict__ W,
                                 const float* __restrict__ bias,
                                 float* __restrict__ Out,
                                 int M, int K, int N, int applySelu) {
    const int lane  = threadIdx.x & 31;
    const int wave  = threadIdx.x >> 5;      // N-tile index
    const int l15   = lane & 15;
    const int half  = lane >> 4;             // 0: K={k,k+1}, 1: K={k+2,k+3}
    const int mBase = blockIdx.x << 4;
    const int nBase = wave << 4;
    const int rowA  = mBase + l15;
    const int col   = nBase + l15;

    const float rs = rowScale ? rowScale[rowA] : 1.0f;

    v8f acc = {};
    const float* aRow = A + (long)rowA * K;
    for (int k = 0; k < K; k += 4) {
        const int ka = k + (half << 1);
        v2f a, b;
        a.x = aRow[ka]     * rs;
        a.y = aRow[ka + 1] * rs;
        b.x = W[(long)ka       * N + col];
        b.y = W[(long)(ka + 1) * N + col];
        acc = __builtin_amdgcn_wmma_f32_16x16x4_f32(
            /*neg_a=*/false, a, /*neg_b=*/false, b,
            /*c_mod=*/(short)0, acc, /*reuse_a=*/false, /*reuse_b=*/false);
    }

    const float bc = bias[col];
#pragma unroll
    for (int r = 0; r < 8; ++r) {
        const int row = mBase + (half << 3) + r;
        float v = acc[r] + bc;
        if (applySelu) v = selu_f(v);
        Out[(long)row * N + col] = v;
    }
}

// ---------------------------------------------------------------- readout
__global__ void readout_kernel(const float* __restrict__ yNode,
                               const int* __restrict__ node2graph,
                               float* __restrict__ ySum) {
    const int chunks = HIDDEN >> 2;  // 32
    const long total = (long)N_NODES * chunks;
    for (long i = blockIdx.x * (long)blockDim.x + threadIdx.x; i < total;
         i += (long)gridDim.x * blockDim.x) {
        const int n = (int)(i / chunks);
        const int c = (int)(i - (long)n * chunks);
        const int g = node2graph[n];
        const float4 v = reinterpret_cast<const float4*>(yNode)[(long)n * chunks + c];
        float* o = ySum + (long)g * HIDDEN + (c << 2);
        atomicAdd(o + 0, v.x);
        atomicAdd(o + 1, v.y);
        atomicAdd(o + 2, v.z);
        atomicAdd(o + 3, v.w);
    }
}

// ------------------------------------------------------------------- MLP
// y1[g,o] = selu( concat(ySum[g], fg[g]) @ L1w + L1b )   (K=131, OUT=256)
__global__ void mlp1_kernel(const float* __restrict__ ySum, const float* __restrict__ fg,
                            const float* __restrict__ L1w, const float* __restrict__ L1b,
                            float* __restrict__ y1) {
    const int OUT = 2 * HIDDEN;
    const long total = (long)N_GRAPHS * OUT;
    for (long i = blockIdx.x * (long)blockDim.x + threadIdx.x; i < total;
         i += (long)gridDim.x * blockDim.x) {
        const int g = (int)(i / OUT);
        const int o = (int)(i - (long)g * OUT);
        float acc = L1b[o];
        const float* row = ySum + (long)g * HIDDEN;
        for (int k = 0; k < HIDDEN; ++k)
            acc += row[k] * L1w[(long)k * OUT + o];
        const float* fgr = fg + (long)g * 3;
        for (int k = 0; k < 3; ++k)
            acc += fgr[k] * L1w[(long)(HIDDEN + k) * OUT + o];
        y1[i] = selu_f(acc);
    }
}

// out[g] = y2[g,:] @ L3w + L3b
__global__ void mlp3_kernel(const float* __restrict__ y2, const float* __restrict__ L3w,
                            const float* __restrict__ L3b, float* __restrict__ out) {
    for (int g = blockIdx.x * blockDim.x + threadIdx.x; g < N_GRAPHS;
         g += gridDim.x * blockDim.x) {
        float acc = L3b[0];
        const float* row = y2 + (long)g * HIDDEN;
        for (int k = 0; k < HIDDEN; ++k)
            acc += row[k] * L3w[k];
        out[g] = acc;
    }
}

// ================================================================== launch
extern "C" void kernel_launch(void* const* d_in, const int* in_sizes, int n_in,
                              void* d_out, int out_size, void* d_ws, size_t ws_size,
                              hipStream_t stream) {
    (void)in_sizes; (void)n_in; (void)out_size; (void)ws_size;

    const float* feats_node = (const float*)d_in[0];
    const float* feats_graph = (const float*)d_in[1];
    const int*   src = (const int*)d_in[2];
    const int*   dst = (const int*)d_in[3];
    const int*   node2graph = (const int*)d_in[4];
    const float* W1 = (const float*)d_in[5];  const float* b1 = (const float*)d_in[6];
    const float* W2 = (const float*)d_in[7];  const float* b2 = (const float*)d_in[8];
    const float* W3 = (const float*)d_in[9];  const float* b3 = (const float*)d_in[10];
    const float* L1w = (const float*)d_in[11]; const float* L1b = (const float*)d_in[12];
    const float* L2w = (const float*)d_in[13]; const float* L2b = (const float*)d_in[14];
    const float* L3w = (const float*)d_in[15]; const float* L3b = (const float*)d_in[16];
    float* out = (float*)d_out;

    // ---- workspace layout (floats), all offsets 16B aligned ----
    float* ws = (float*)d_ws;
    float* normOut = ws;                            // N_NODES
    float* normIn  = normOut + N_NODES;             // N_NODES
    float* bufA    = normIn + N_NODES;              // N_NODES*128
    float* bufB    = bufA + (long)N_NODES * HIDDEN; // N_NODES*128
    float* ySum    = bufB + (long)N_NODES * HIDDEN; // N_GRAPHS*128
    float* y1      = ySum + (long)N_GRAPHS * HIDDEN;     // N_GRAPHS*256
    float* y2      = y1 + (long)N_GRAPHS * 2 * HIDDEN;   // N_GRAPHS*128

    const int ZB = 256;
    const int gridBig = 4096;

    // 1) degrees -> norms
    zero_kernel<<<512, ZB, 0, stream>>>(normOut, N_NODES);
    zero_kernel<<<512, ZB, 0, stream>>>(normIn, N_NODES);
    degree_kernel<<<gridBig, ZB, 0, stream>>>(src, dst, normOut, normIn, N_EDGES);
    finalize_norm_kernel<<<512, ZB, 0, stream>>>(normOut, N_NODES);
    finalize_norm_kernel<<<512, ZB, 0, stream>>>(normIn, N_NODES);

    const int mTiles = N_NODES / 16;            // 6250
    const int gemmBlk = (HIDDEN / 16) * 32;     // 8 waves = 256 threads

    // 2) layer 1: agg(feats_node, F=64) -> bufA ; gemm K=64 -> bufB (selu)
    zero_kernel<<<gridBig, ZB, 0, stream>>>(bufA, (long)N_NODES * IN_FEATS);
    edge_agg_kernel<<<gridBig, ZB, 0, stream>>>(feats_node, normOut, src, dst, bufA,
                                                N_EDGES, IN_FEATS);
    gemm_wmma_kernel<<<mTiles, gemmBlk, 0, stream>>>(bufA, normIn, W1, b1, bufB,
                                                     N_NODES, IN_FEATS, HIDDEN, 1);

    // 3) layer 2
    zero_kernel<<<gridBig, ZB, 0, stream>>>(bufA, (long)N_NODES * HIDDEN);
    edge_agg_kernel<<<gridBig, ZB, 0, stream>>>(bufB, normOut, src, dst, bufA,
                                                N_EDGES, HIDDEN);
    gemm_wmma_kernel<<<mTiles, gemmBlk, 0, stream>>>(bufA, normIn, W2, b2, bufB,
                                                     N_NODES, HIDDEN, HIDDEN, 1);

    // 4) layer 3 (no selu) -> y_node in bufB
    zero_kernel<<<gridBig, ZB, 0, stream>>>(bufA, (long)N_NODES * HIDDEN);
    edge_agg_kernel<<<gridBig, ZB, 0, stream>>>(bufB, normOut, src, dst, bufA,
                                                N_EDGES, HIDDEN);
    gemm_wmma_kernel<<<mTiles, gemmBlk, 0, stream>>>(bufA, normIn, W3, b3, bufB,
                                                     N_NODES, HIDDEN, HIDDEN, 0);

    // 5) readout
    zero_kernel<<<512, ZB, 0, stream>>>(ySum, (long)N_GRAPHS * HIDDEN);
    readout_kernel<<<gridBig, ZB, 0, stream>>>(bufB, node2graph, ySum);

    // 6) MLP: L1 scalar (K=131), L2 via WMMA (4096x256 @ 256x128), L3 scalar
    mlp1_kernel<<<2048, ZB, 0, stream>>>(ySum, feats_graph, L1w, L1b, y1);
    gemm_wmma_kernel<<<N_GRAPHS / 16, gemmBlk, 0, stream>>>(y1, nullptr, L2w, L2b, y2,
                                                            N_GRAPHS, 2 * HIDDEN, HIDDEN, 1);
    mlp3_kernel<<<16, ZB, 0, stream>>>(y2, L3w, L3b, out);
}